// HDLoss_85409719648670
// MI455X (gfx1250) — compile-verified
//
#include <hip/hip_runtime.h>
#include <stdint.h>

// Problem constants (from reference setup_inputs).
#define BB 16
#define HH 512
#define WW 512
#define HWPIX (HH * WW)
#define NPIX (BB * HWPIX)
#define LARGE 524289.0f  // H*H + W*W + 1

typedef __attribute__((ext_vector_type(2))) float v2f;
typedef __attribute__((ext_vector_type(8))) float v8f;

// ---------------------------------------------------------------------------
// Kernel 1: per-column pass.
//  - masks: pc = (o1 > o0)  [sigmoid(o1-o0) > 0.5], gt = (target == 1)
//  - perr  = (sigmoid(o1-o0) - gt)^2                       (channel-1 error)
//  - f_*   = min(d1^2, LARGE), d1 = 1-D run distance along H (fwd/bwd scans)
// One thread per (b, w) column; lanes stride w -> coalesced per h step.
// ---------------------------------------------------------------------------
__global__ void hd_col_pass(const float* __restrict__ logits,
                            const int* __restrict__ tgt,
                            float* __restrict__ perr,
                            float* __restrict__ fpc,
                            float* __restrict__ fgt) {
    int col = blockIdx.x * blockDim.x + threadIdx.x;
    if (col >= BB * WW) return;
    int b = col / WW;
    int w = col - b * WW;

    const float* o0p = logits + (size_t)b * 2 * HWPIX + w;       // channel 0
    const float* o1p = o0p + HWPIX;                              // channel 1
    const int*   tp  = tgt  + (size_t)b * HWPIX + w;
    float*       pe  = perr + (size_t)b * HWPIX + w;
    float*       fp  = fpc  + (size_t)b * HWPIX + w;
    float*       fg  = fgt  + (size_t)b * HWPIX + w;

    // forward scan (store running distance)
    float cp = LARGE, cg = LARGE;
    for (int h = 0; h < HH; ++h) {
        int off = h * WW;
        float o0 = o0p[off], o1 = o1p[off];
        bool mp = o1 > o0;
        bool mg = (tp[off] == 1);
        cp = mp ? cp + 1.0f : 0.0f;
        cg = mg ? cg + 1.0f : 0.0f;
        fp[off] = cp;
        fg[off] = cg;
        float p = 1.0f / (1.0f + __expf(o0 - o1));  // softmax channel 1
        float e = p - (mg ? 1.0f : 0.0f);
        pe[off] = e * e;
    }
    // backward scan: combine, square, cap
    cp = LARGE; cg = LARGE;
    for (int h = HH - 1; h >= 0; --h) {
        int off = h * WW;
        float o0 = o0p[off], o1 = o1p[off];
        bool mp = o1 > o0;
        bool mg = (tp[off] == 1);
        cp = mp ? cp + 1.0f : 0.0f;
        cg = mg ? cg + 1.0f : 0.0f;
        float dp = fminf(fp[off], cp);
        float dg = fminf(fg[off], cg);
        fp[off] = fminf(dp * dp, LARGE);
        fg[off] = fminf(dg * dg, LARGE);
    }
}

// ---------------------------------------------------------------------------
// Kernel 2: per-row lower-envelope pass, fused with the loss weighting.
// One wave (32 lanes) per row; 8 rows per 256-thread block; 1024 blocks.
// f rows are staged into LDS with gfx1250 async global->LDS loads
// (ASYNCcnt path), then each lane owns 16 j positions for the branchless
// D[j] = min_k f[k] + (j-k)^2 sweep over both masks simultaneously.
// ---------------------------------------------------------------------------
__global__ void hd_row_pass(const float* __restrict__ perr,
                            const float* __restrict__ fpc,
                            const float* __restrict__ fgt,
                            float* __restrict__ partial) {
    __shared__ float sf[8][2][WW];   // 8 rows x {pc,gt} x 512 f32 = 32 KB
    __shared__ float wsum[8];

    int lane = threadIdx.x & 31;
    int wid  = threadIdx.x >> 5;
    int row  = blockIdx.x * 8 + wid;            // row index in [0, B*H)
    size_t rowbase = (size_t)row * WW;

    // LDS byte offsets (low 32 bits of the flat shared pointer = LDS offset)
    unsigned ldsp = (unsigned)(uintptr_t)&sf[wid][0][0];
    unsigned ldsg = (unsigned)(uintptr_t)&sf[wid][1][0];
    const float* gp = fpc + rowbase;
    const float* gg = fgt + rowbase;

    // Stage both 2 KB rows: 4 x b128 issues per row, 16 B per lane per issue.
#pragma unroll
    for (int it = 0; it < 4; ++it) {
        unsigned lofs = (unsigned)(it * 512) + (unsigned)lane * 16u;
        uint64_t ap = (uint64_t)(uintptr_t)(gp + it * 128 + lane * 4);
        uint64_t ag = (uint64_t)(uintptr_t)(gg + it * 128 + lane * 4);
        unsigned dpa = ldsp + lofs;
        unsigned dga = ldsg + lofs;
        asm volatile("global_load_async_to_lds_b128 %0, %1, off"
                     :: "v"(dpa), "v"(ap) : "memory");
        asm volatile("global_load_async_to_lds_b128 %0, %1, off"
                     :: "v"(dga), "v"(ag) : "memory");
    }
    asm volatile("s_wait_asynccnt 0x0" ::: "memory");

    float Dp[16], Dg[16], jv[16];
#pragma unroll
    for (int jj = 0; jj < 16; ++jj) {
        Dp[jj] = LARGE;
        Dg[jj] = LARGE;
        jv[jj] = (float)(jj * 32 + lane);
    }
    for (int k = 0; k < WW; ++k) {
        float fkp = sf[wid][0][k];   // LDS broadcast (uniform address)
        float fkg = sf[wid][1][k];
        float kf  = (float)k;
#pragma unroll
        for (int jj = 0; jj < 16; ++jj) {
            float t  = jv[jj] - kf;
            float tt = t * t;
            Dp[jj] = fminf(Dp[jj], fkp + tt);
            Dg[jj] = fminf(Dg[jj], fkg + tt);
        }
    }

    // fuse the loss weighting and reduce
    float local = 0.0f;
#pragma unroll
    for (int jj = 0; jj < 16; ++jj) {
        int j = jj * 32 + lane;
        local += perr[rowbase + j] * (Dp[jj] + Dg[jj]);
    }
    for (int off = 16; off > 0; off >>= 1)
        local += __shfl_xor(local, off, 32);
    if (lane == 0) wsum[wid] = local;
    __syncthreads();
    if (threadIdx.x == 0) {
        float s = 0.0f;
        for (int i = 0; i < 8; ++i) s += wsum[i];
        partial[blockIdx.x] = s;
    }
}

// ---------------------------------------------------------------------------
// Kernel 3: deterministic final reduction of 1024 block partials using the
// f32 WMMA as an exact summation engine: with B = ones(4x16),
// D[m][n] = sum_k A[m][k] + C[m][n], so accumulating C over 16 tiles and
// summing all 256 C entries yields 16 * total (each column repeats the sum).
// Single wave of 32 lanes; EXEC is all-1s around the WMMA as required.
// ---------------------------------------------------------------------------
__global__ void hd_final_reduce(const float* __restrict__ partial,
                                float* __restrict__ outp) {
    int lane = threadIdx.x;          // 0..31, launched with exactly 32 threads
    v8f c = {};
    v2f ones;
    ones.x = 1.0f;
    ones.y = 1.0f;
    for (int it = 0; it < 1024; it += 64) {
        v2f a;
        a.x = partial[it + lane * 2 + 0];
        a.y = partial[it + lane * 2 + 1];
        // (neg_a, A, neg_b, B, c_mod, C, reuse_a, reuse_b)
        c = __builtin_amdgcn_wmma_f32_16x16x4_f32(false, a, false, ones,
                                                  (short)0, c, false, false);
    }
    float s = 0.0f;
#pragma unroll
    for (int i = 0; i < 8; ++i) s += c[i];
    for (int off = 16; off > 0; off >>= 1)
        s += __shfl_xor(s, off, 32);
    if (lane == 0)
        outp[0] = s * (1.0f / 16.0f) * (1.0f / (float)NPIX);
}

// ---------------------------------------------------------------------------
extern "C" void kernel_launch(void* const* d_in, const int* in_sizes, int n_in,
                              void* d_out, int out_size, void* d_ws, size_t ws_size,
                              hipStream_t stream) {
    (void)in_sizes; (void)n_in; (void)out_size; (void)ws_size;
    const float* logits = (const float*)d_in[0];   // (16,2,512,512) f32
    const int*   tgt    = (const int*)d_in[1];     // (16,512,512) int

    // workspace layout: perr | f_pc | f_gt | partials   (~50.4 MB)
    float* perr    = (float*)d_ws;
    float* fpc     = perr + NPIX;
    float* fgt     = fpc + NPIX;
    float* partial = fgt + NPIX;                   // 1024 floats

    hd_col_pass<<<(BB * WW + 255) / 256, 256, 0, stream>>>(logits, tgt, perr,
                                                           fpc, fgt);
    hd_row_pass<<<(BB * HH) / 8, 256, 0, stream>>>(perr, fpc, fgt, partial);
    hd_final_reduce<<<1, 32, 0, stream>>>(partial, (float*)d_out);
}